// MambaBlock_59072980189822
// MI455X (gfx1250) — compile-verified
//
#include <hip/hip_runtime.h>
#include <math.h>

// ---------------- constants from the reference ----------------
#define DIMC     256
#define STEPC    128
#define D_INNER  1024
#define D_STATE  16
#define D_CONV   4
#define DT_RANK  16
#define B_SZ_C   4
#define T_LEN_C  131072
#define L_FRAMES 1023           // (T_LEN - DIM)/STEP + 1
#define EPSC     1e-5f

typedef float v2f __attribute__((ext_vector_type(2)));
typedef float v8f __attribute__((ext_vector_type(8)));

__device__ __forceinline__ float silu_f(float x) {
    return x / (1.0f + __expf(-x));
}
__device__ __forceinline__ float softplus_f(float x) {
    return (x > 20.0f) ? x : log1pf(__expf(x));
}

// ---------------- frames: h[b,l,j] = x[b, l*128 + j] ----------------
__global__ void frames_kernel(const float* __restrict__ x, float* __restrict__ h) {
    size_t gid = (size_t)blockIdx.x * blockDim.x + threadIdx.x;
    size_t total = (size_t)B_SZ_C * L_FRAMES * DIMC;
    if (gid >= total) return;
    int j = gid % DIMC;
    size_t tok = gid / DIMC;
    int l = tok % L_FRAMES;
    int b = tok / L_FRAMES;
    h[gid] = x[(size_t)b * T_LEN_C + (size_t)l * STEPC + j];
}

// ---------------- residual add + RMSNorm (one block of 256 per token) -------
template <bool FIRST>
__global__ void add_norm_kernel(const float* __restrict__ h,
                                float* __restrict__ res,
                                float* __restrict__ hn,
                                const float* __restrict__ nw) {
    size_t tok = blockIdx.x;
    int j = threadIdx.x;
    size_t idx = tok * DIMC + j;
    float r = FIRST ? h[idx] : (res[idx] + h[idx]);
    res[idx] = r;
    __shared__ float s[DIMC];
    s[j] = r * r;
    __syncthreads();
    for (int o = DIMC / 2; o > 0; o >>= 1) {
        if (j < o) s[j] += s[j + o];
        __syncthreads();
    }
    float scale = rsqrtf(s[0] * (1.0f / DIMC) + EPSC);
    hn[idx] = r * scale * nw[j];
}

// ---------------- fp32 WMMA GEMM: C[m,e] = sum_k A[m,k] * W[e,k] ------------
// One wave computes a 16x16 tile of C using v_wmma_f32_16x16x4_f32.
// A-frag (16x4 f32): lane<16 -> {K=k, k+1}, lane>=16 -> {K=k+2, k+3}, M = lane&15
// B-frag (4x16 f32): lane's 2 VGPRs hold rows K=kb, kb+1 for col N=lane&15
// C/D  (16x16 f32): VGPR v: lanes 0-15 -> M=v, lanes 16-31 -> M=v+8, N=lane&15
//
// Edge handling: A row index is CLAMPED (not masked) so all loads are
// unconditionally in-bounds and the inner loop is branch-free (keeps EXEC
// all-1s, no saveexec churn). Out-of-range rows compute garbage that is
// simply never stored.
// EPI: 0 = none, 1 = softplus(acc + bias[e])
template <int EPI>
__global__ void gemm_wmma_f32(const float* __restrict__ A, int lda,
                              const float* __restrict__ W,      // (E x K) row-major
                              const float* __restrict__ bias,   // [E] or null
                              float* __restrict__ C, int ldc,
                              int M, int E, int K) {
    int gthread = blockIdx.x * blockDim.x + threadIdx.x;
    int wave = gthread >> 5;
    int lane = threadIdx.x & 31;
    int tilesE = E >> 4;
    int totalWaves = ((M + 15) >> 4) * tilesE;
    if (wave >= totalWaves) return;            // wave-uniform: EXEC stays all-1s

    int tM = (wave / tilesE) << 4;
    int tE = (wave % tilesE) << 4;

    int half2 = (lane >> 4) << 1;              // 0 or 2
    int mrow = tM + (lane & 15);
    int ecol = tE + (lane & 15);
    int mr = (mrow < M) ? mrow : (M - 1);      // clamp: loads always in-bounds

    const float* __restrict__ arow = A + (size_t)mr * lda + half2;
    const float* __restrict__ wrow = W + (size_t)ecol * K + half2;

    v8f acc = {};
    // software-pipelined, branch-free K loop
    v2f a0, b0;
    a0.x = arow[0]; a0.y = arow[1];
    b0.x = wrow[0]; b0.y = wrow[1];
#pragma unroll 4
    for (int k = 4; k < K; k += 4) {
        v2f a1, b1;
        a1.x = arow[k];     a1.y = arow[k + 1];
        b1.x = wrow[k];     b1.y = wrow[k + 1];
        acc = __builtin_amdgcn_wmma_f32_16x16x4_f32(
            false, a0, false, b0, (short)0, acc, false, false);
        a0 = a1; b0 = b1;
    }
    acc = __builtin_amdgcn_wmma_f32_16x16x4_f32(
        false, a0, false, b0, (short)0, acc, false, false);

#pragma unroll
    for (int v = 0; v < 8; ++v) {
        int m = tM + v + ((lane >> 4) << 3);
        if (m < M) {
            float val = acc[v];
            if (EPI == 1) val = softplus_f(val + bias[ecol]);
            C[(size_t)m * ldc + ecol] = val;
        }
    }
}

// ---------------- causal depthwise conv(4) + bias + SiLU --------------------
// input: xc-half of xz (row stride 2048), output: xc buffer (row stride 1024)
__global__ void conv_silu_kernel(const float* __restrict__ xz,
                                 const float* __restrict__ cw,   // (1024,4)
                                 const float* __restrict__ cb,   // (1024)
                                 float* __restrict__ xc) {
    size_t gid = (size_t)blockIdx.x * blockDim.x + threadIdx.x;
    size_t total = (size_t)B_SZ_C * L_FRAMES * D_INNER;
    if (gid >= total) return;
    int d = gid % D_INNER;
    size_t tok = gid / D_INNER;
    int l = tok % L_FRAMES;
    int b = tok / L_FRAMES;
    float acc = cb[d];
#pragma unroll
    for (int j = 0; j < D_CONV; ++j) {
        int ls = l - (D_CONV - 1) + j;
        if (ls >= 0)
            acc += cw[d * D_CONV + j] *
                   xz[((size_t)b * L_FRAMES + ls) * 2048 + d];
    }
    xc[tok * D_INNER + d] = silu_f(acc);
}

// ---------------- selective scan: thread per (b, d, n) state ----------------
// 65536 threads (2048 waves); h*C reduced over 16 state lanes via shfl_xor.
// Fuses y = (scan + u*D) * silu(z); writes y into xc-half of xz buffer.
__global__ void scan_kernel(const float* __restrict__ xc,     // u, N x 1024
                            const float* __restrict__ delta,  // N x 1024
                            const float* __restrict__ xdbl,   // N x 48 (dt|B|C)
                            const float* __restrict__ A_log,  // 1024 x 16
                            const float* __restrict__ Dp,     // 1024
                            const float* __restrict__ xz,     // z at +1024, ld 2048
                            float* __restrict__ y) {          // ld 2048
    int gid = blockIdx.x * blockDim.x + threadIdx.x;
    if (gid >= B_SZ_C * D_INNER * D_STATE) return;
    int n = gid & 15;
    int d = (gid >> 4) & (D_INNER - 1);
    int b = gid >> 14;

    float An = -__expf(A_log[d * D_STATE + n]);
    float Dd = Dp[d];
    float h = 0.0f;

    for (int l = 0; l < L_FRAMES; ++l) {
        size_t tok = (size_t)b * L_FRAMES + l;
        float u  = xc[tok * D_INNER + d];
        float dl = delta[tok * D_INNER + d];
        float Bn = xdbl[tok * 48 + DT_RANK + n];
        float Cn = xdbl[tok * 48 + DT_RANK + D_STATE + n];
        h = __expf(dl * An) * h + (dl * u) * Bn;
        float p = h * Cn;
        p += __shfl_xor(p, 1, 16);
        p += __shfl_xor(p, 2, 16);
        p += __shfl_xor(p, 4, 16);
        p += __shfl_xor(p, 8, 16);
        if (n == 0) {
            float z = xz[tok * 2048 + D_INNER + d];
            y[tok * 2048 + d] = (p + u * Dd) * silu_f(z);
        }
    }
}

// ---------------- overlap-add fold: out[b,t] = x + sum of windowed frames ---
__global__ void fold_kernel(const float* __restrict__ x,
                            const float* __restrict__ h,
                            const float* __restrict__ res,
                            float* __restrict__ out) {
    size_t gid = (size_t)blockIdx.x * blockDim.x + threadIdx.x;
    if (gid >= (size_t)B_SZ_C * T_LEN_C) return;
    int b = gid / T_LEN_C;
    int t = gid % T_LEN_C;
    float acc = x[gid];
    int l1 = t >> 7;           // frame with j in [0,128)
    int j1 = t & 127;
    if (l1 < L_FRAMES) {
        size_t i = ((size_t)b * L_FRAMES + l1) * DIMC + j1;
        float w = (float)j1 * (1.0f / 127.0f);
        acc += (h[i] + res[i]) * w;
    }
    if (l1 >= 1) {             // previous frame, j in [128,256)
        int l0 = l1 - 1, j0 = j1 + 128;
        size_t i = ((size_t)b * L_FRAMES + l0) * DIMC + j0;
        float w = (float)(255 - j0) * (1.0f / 127.0f);
        acc += (h[i] + res[i]) * w;
    }
    out[gid] = acc;
}

// ---------------- host side -------------------------------------------------
template <int EPI>
static void launch_gemm(const float* A, int lda, const float* W, const float* bias,
                        float* C, int ldc, int M, int E, int K, hipStream_t s) {
    long waves = (long)((M + 15) / 16) * (E / 16);
    long threads = waves * 32;
    int blocks = (int)((threads + 255) / 256);
    gemm_wmma_f32<EPI><<<blocks, 256, 0, s>>>(A, lda, W, bias, C, ldc, M, E, K);
}

extern "C" void kernel_launch(void* const* d_in, const int* in_sizes, int n_in,
                              void* d_out, int out_size, void* d_ws, size_t ws_size,
                              hipStream_t stream) {
    const float* x      = (const float*)d_in[0];
    const float* in_w   = (const float*)d_in[1];   // (2, 2048, 256)
    const float* conv_w = (const float*)d_in[2];   // (2, 1024, 1, 4)
    const float* conv_b = (const float*)d_in[3];   // (2, 1024)
    const float* xp_w   = (const float*)d_in[4];   // (2, 48, 1024)
    const float* dt_w   = (const float*)d_in[5];   // (2, 1024, 16)
    const float* dt_b   = (const float*)d_in[6];   // (2, 1024)
    const float* A_log  = (const float*)d_in[7];   // (2, 1024, 16)
    const float* Dp     = (const float*)d_in[8];   // (2, 1024)
    const float* out_w  = (const float*)d_in[9];   // (2, 256, 1024)
    const float* norm_w = (const float*)d_in[10];  // (2, 256)
    float* out = (float*)d_out;

    const size_t N = (size_t)B_SZ_C * L_FRAMES;   // 4092 tokens

    float* ws  = (float*)d_ws;
    float* bRes = ws;  ws += N * DIMC;            // residual accumulator
    float* bHn  = ws;  ws += N * DIMC;            // rmsnorm output
    float* bH   = ws;  ws += N * DIMC;            // current layer input/output
    float* bXZ  = ws;  ws += N * 2048;            // xz (later y | z)
    float* bXC  = ws;  ws += N * D_INNER;         // conv+silu output (u)
    float* bXD  = ws;  ws += N * 48;              // x_dbl (dt | B | C)
    float* bDl  = ws;  ws += N * D_INNER;         // delta

    {
        size_t tot = N * DIMC;
        frames_kernel<<<(int)((tot + 255) / 256), 256, 0, stream>>>(x, bH);
    }

    for (int i = 0; i < 2; ++i) {
        const float* nw_i  = norm_w + (size_t)i * DIMC;
        const float* inw_i = in_w   + (size_t)i * 2 * D_INNER * DIMC;
        const float* cw_i  = conv_w + (size_t)i * D_INNER * D_CONV;
        const float* cb_i  = conv_b + (size_t)i * D_INNER;
        const float* xpw_i = xp_w   + (size_t)i * 48 * D_INNER;
        const float* dtw_i = dt_w   + (size_t)i * D_INNER * DT_RANK;
        const float* dtb_i = dt_b   + (size_t)i * D_INNER;
        const float* Al_i  = A_log  + (size_t)i * D_INNER * D_STATE;
        const float* Dp_i  = Dp     + (size_t)i * D_INNER;
        const float* ow_i  = out_w  + (size_t)i * DIMC * D_INNER;

        if (i == 0)
            add_norm_kernel<true><<<(int)N, DIMC, 0, stream>>>(bH, bRes, bHn, nw_i);
        else
            add_norm_kernel<false><<<(int)N, DIMC, 0, stream>>>(bH, bRes, bHn, nw_i);

        // in_proj: (N x 256) x (2048 x 256)^T -> xz (N x 2048)
        launch_gemm<0>(bHn, DIMC, inw_i, nullptr, bXZ, 2048,
                       (int)N, 2 * D_INNER, DIMC, stream);

        {
            size_t tot = N * D_INNER;
            conv_silu_kernel<<<(int)((tot + 255) / 256), 256, 0, stream>>>(
                bXZ, cw_i, cb_i, bXC);
        }

        // x_proj: (N x 1024) x (48 x 1024)^T -> x_dbl (N x 48)
        launch_gemm<0>(bXC, D_INNER, xpw_i, nullptr, bXD, 48,
                       (int)N, 48, D_INNER, stream);

        // dt_proj + softplus: (N x 16) x (1024 x 16)^T -> delta (N x 1024)
        launch_gemm<1>(bXD, 48, dtw_i, dtb_i, bDl, D_INNER,
                       (int)N, D_INNER, DT_RANK, stream);

        // selective scan + D skip + silu(z) gate; y written into xc-half of bXZ
        scan_kernel<<<(B_SZ_C * D_INNER * D_STATE) / 256, 256, 0, stream>>>(
            bXC, bDl, bXD, Al_i, Dp_i, bXZ, bXZ);

        // out_proj: (N x 1024, ld 2048) x (256 x 1024)^T -> h (N x 256)
        launch_gemm<0>(bXZ, 2048, ow_i, nullptr, bH, DIMC,
                       (int)N, DIMC, D_INNER, stream);
    }

    {
        size_t tot = (size_t)B_SZ_C * T_LEN_C;
        fold_kernel<<<(int)((tot + 255) / 256), 256, 0, stream>>>(x, bH, bRes, out);
    }
}